// DyNet2D_31155692765576
// MI455X (gfx1250) — compile-verified
//
#include <hip/hip_runtime.h>

typedef float v2f __attribute__((ext_vector_type(2)));
typedef float v8f __attribute__((ext_vector_type(8)));

#define BSZ   16
#define CIN   16
#define COUT  16
#define HH    512
#define WW    512
#define KDIM  144          // 9 positions * 16 input channels
#define WSTR  146          // LDS weight row stride (even -> 8B aligned b64, conflict-free)
#define XSTR  34           // LDS x-tile row stride (32 pixels + 2 halo)
#define SPLITS 32          // pooling partial splits per (b,c)

__device__ __forceinline__ int reflect_idx(int v, int n) {
    v = (v < 0) ? -v : v;
    v = (v >= n) ? (2 * n - 2 - v) : v;
    return v;
}

// ---------------- Phase 1: pooling partial sums (deterministic) ----------------
// NT loads: this pass streams all 256MB of x once; don't pollute L2 for the conv pass.
__global__ __launch_bounds__(256) void pool_partial_kernel(const float* __restrict__ x,
                                                           float* __restrict__ part) {
    const int blk = blockIdx.x;              // (b*16+c)*SPLITS + s
    const int bc  = blk / SPLITS;
    const int s   = blk % SPLITS;
    const int chunk = (HH * WW) / SPLITS;    // 8192
    const float* p = x + (size_t)bc * (HH * WW) + (size_t)s * chunk;

    float sum = 0.0f;
    for (int j = threadIdx.x; j < chunk; j += 256)
        sum += __builtin_nontemporal_load(&p[j]);

    __shared__ float red[256];
    red[threadIdx.x] = sum;
    __syncthreads();
    for (int off = 128; off > 0; off >>= 1) {
        if (threadIdx.x < off) red[threadIdx.x] += red[threadIdx.x + off];
        __syncthreads();
    }
    if (threadIdx.x == 0) part[blk] = red[0];
}

// ---------------- Phase 2: finish pooling, softmax routing, mix expert kernels ----------------
// wmix layout: [b][o][k], k = (kh*3+kw)*16 + i   (K-quads stay inside one (kh,kw) position)
__global__ __launch_bounds__(256) void route_mix_kernel(const float* __restrict__ part,
                                                        const float* __restrict__ w_experts,
                                                        const float* __restrict__ fc_w,
                                                        const float* __restrict__ fc_b,
                                                        float* __restrict__ wmix) {
    __shared__ float pooled[BSZ][CIN];
    __shared__ float logit[BSZ][3];
    __shared__ float rout[BSZ][3];
    const int tid = threadIdx.x;

    {   // 256 threads == 16*16 (b,c) pairs
        const int b = tid >> 4, c = tid & 15;
        float s = 0.0f;
        for (int j = 0; j < SPLITS; ++j) s += part[(b * CIN + c) * SPLITS + j];
        pooled[b][c] = s * (1.0f / (float)(HH * WW));
    }
    __syncthreads();

    if (tid < BSZ * 3) {
        const int b = tid / 3, e = tid % 3;
        float s = fc_b[e];
        for (int c = 0; c < CIN; ++c) s += pooled[b][c] * fc_w[e * CIN + c];
        logit[b][e] = s;
    }
    __syncthreads();

    if (tid < BSZ) {
        const float m  = fmaxf(logit[tid][0], fmaxf(logit[tid][1], logit[tid][2]));
        const float e0 = expf(logit[tid][0] - m);
        const float e1 = expf(logit[tid][1] - m);
        const float e2 = expf(logit[tid][2] - m);
        const float inv = 1.0f / (e0 + e1 + e2);
        rout[tid][0] = e0 * inv; rout[tid][1] = e1 * inv; rout[tid][2] = e2 * inv;
    }
    __syncthreads();

    for (int e = tid; e < BSZ * COUT * KDIM; e += 256) {
        const int b   = e / (COUT * KDIM);
        const int rem = e % (COUT * KDIM);
        const int o   = rem / KDIM;
        const int k   = rem % KDIM;
        const int p   = k >> 4;          // kh*3+kw
        const int i   = k & 15;          // input channel
        float v = 0.0f;
        #pragma unroll
        for (int ex = 0; ex < 3; ++ex)
            v += rout[b][ex] * w_experts[((ex * COUT + o) * CIN + i) * 9 + p];
        wmix[e] = v;
    }
}

// ---------------- Phase 3: implicit-GEMM conv via V_WMMA_F32_16X16X4_F32 ----------------
// Block: 256 thr = 8 waves; tile: 8 rows x 32 cols of one sample, all 16 out channels.
// Per wave: two adjacent D[16 outch x 16 px] tiles sharing each A operand.
// 72 WMMAs/wave across 4 independent accumulator chains.
__global__ __launch_bounds__(256) void conv_wmma_kernel(const float* __restrict__ x,
                                                        const float* __restrict__ wmix,
                                                        const float* __restrict__ bias,
                                                        float* __restrict__ out) {
    const int x0 = blockIdx.x * 32;
    const int y0 = blockIdx.y * 8;
    const int b  = blockIdx.z;

    __shared__ float sX[CIN * 10 * XSTR];  // x tile + reflect halo: [i][row 0..9][col 0..33]
    __shared__ float sW[COUT * WSTR];      // mixed weights: [o][k], row stride 146

    // cooperative x-tile load with reflect padding (RT: halos shared with neighbor blocks)
    for (int e = threadIdx.x; e < CIN * 10 * XSTR; e += 256) {
        const int i   = e / (10 * XSTR);
        const int rem = e % (10 * XSTR);
        const int rr  = rem / XSTR;
        const int cc  = rem % XSTR;
        const int gy  = reflect_idx(y0 + rr - 1, HH);
        const int gx  = reflect_idx(x0 + cc - 1, WW);
        sX[e] = x[(((size_t)b * CIN + i) * HH + gy) * WW + gx];
    }
    // cooperative weight load
    for (int e = threadIdx.x; e < COUT * KDIM; e += 256) {
        const int o = e / KDIM, k = e % KDIM;
        sW[o * WSTR + k] = wmix[((size_t)b * COUT + o) * KDIM + k];
    }
    __syncthreads();

    const int lane = threadIdx.x & 31;
    const int wid  = threadIdx.x >> 5;    // tile row 0..7
    const int h    = lane >> 4;           // lane half (K-pair / M-high select)
    const int n    = lane & 15;           // A: row M; B: pixel column N

    v8f acc0 = {0.f,0.f,0.f,0.f,0.f,0.f,0.f,0.f};   // pixel tile 0, even p
    v8f acc1 = {0.f,0.f,0.f,0.f,0.f,0.f,0.f,0.f};   // pixel tile 0, odd p
    v8f acc2 = {0.f,0.f,0.f,0.f,0.f,0.f,0.f,0.f};   // pixel tile 1, even p
    v8f acc3 = {0.f,0.f,0.f,0.f,0.f,0.f,0.f,0.f};   // pixel tile 1, odd p

    #pragma unroll
    for (int p = 0; p < 9; ++p) {
        const int kh = p / 3, kw = p % 3;
        const int r  = wid + kh;                      // LDS row within halo tile
        #pragma unroll
        for (int c = 0; c < 4; ++c) {
            const int k0 = p * 16 + c * 4;            // K-quad base
            // A 16x4 f32 layout: lane m=n; VGPR0/1 hold K = k0+2h, k0+2h+1 (shared by both tiles)
            const v2f a = *(const v2f*)&sW[n * WSTR + k0 + 2 * h];
            // B 4x16 f32 layout: VGPR0 rows K{0,2}, VGPR1 rows K{1,3} across lane halves
            const int i0  = c * 4 + 2 * h;            // input channel for this lane's K rows
            const int ro0 = (i0 * 10 + r) * XSTR + n + kw;
            const int ro1 = ((i0 + 1) * 10 + r) * XSTR + n + kw;
            v2f b0, b1;
            b0.x = sX[ro0];        b0.y = sX[ro1];         // pixel tile 0
            b1.x = sX[ro0 + 16];   b1.y = sX[ro1 + 16];    // pixel tile 1
            if ((p & 1) == 0) {
                acc0 = __builtin_amdgcn_wmma_f32_16x16x4_f32(false, a, false, b0,
                                                             (short)0, acc0, false, false);
                acc2 = __builtin_amdgcn_wmma_f32_16x16x4_f32(false, a, false, b1,
                                                             (short)0, acc2, false, false);
            } else {
                acc1 = __builtin_amdgcn_wmma_f32_16x16x4_f32(false, a, false, b0,
                                                             (short)0, acc1, false, false);
                acc3 = __builtin_amdgcn_wmma_f32_16x16x4_f32(false, a, false, b1,
                                                             (short)0, acc3, false, false);
            }
        }
    }

    // D layout: vgpr v -> out channel M = v + 8h, lane -> pixel N. NT stores: pure streaming output.
    const int y = y0 + wid;
    #pragma unroll
    for (int v = 0; v < 8; ++v) {
        const int o = v + 8 * h;
        float* rowp = &out[(((size_t)b * COUT + o) * HH + y) * WW + x0 + n];
        __builtin_nontemporal_store(acc0[v] + acc1[v] + bias[o], rowp);
        __builtin_nontemporal_store(acc2[v] + acc3[v] + bias[o], rowp + 16);
    }
}

extern "C" void kernel_launch(void* const* d_in, const int* in_sizes, int n_in,
                              void* d_out, int out_size, void* d_ws, size_t ws_size,
                              hipStream_t stream) {
    const float* x     = (const float*)d_in[0];  // [16,16,512,512]
    const float* wexp  = (const float*)d_in[1];  // [3,16,16,3,3]
    const float* bias  = (const float*)d_in[2];  // [16]
    const float* fc_w  = (const float*)d_in[3];  // [3,16]
    const float* fc_b  = (const float*)d_in[4];  // [3]
    float* out = (float*)d_out;

    float* wmix = (float*)d_ws;                  // 16*16*144 = 36864 floats
    float* part = wmix + BSZ * COUT * KDIM;      // 16*16*32  =  8192 floats

    pool_partial_kernel<<<BSZ * CIN * SPLITS, 256, 0, stream>>>(x, part);
    route_mix_kernel<<<1, 256, 0, stream>>>(part, wexp, fc_w, fc_b, wmix);

    dim3 grid(WW / 32, HH / 8, BSZ);
    conv_wmma_kernel<<<grid, 256, 0, stream>>>(x, wmix, bias, out);
}